// SPDRectified_81140522156468
// MI455X (gfx1250) — compile-verified
//
#include <hip/hip_runtime.h>

// SPDRectified: out = U max(Λ, eps) U^T for SPD X = U Λ U^T.
// Rewritten GEMM-only via matrix sign function:
//   Y = X - eps*I;  |Y| = sign(Y)*Y;  out = 0.5*(Y + |Y|) + eps*I
// sign(Y) by Newton-Schulz: Z <- 1.5 Z - 0.5 Z^3, Z0 = Y/||Y||_F.
// All 64x64 GEMMs run on V_WMMA_F32_16X16X4_F32 (fp32 matrix pipe; f16 would
// not survive 28 sign iterations with eigenvalues spanning 1e-4..4).

#define EPS       1e-4f
#define LD        68          // LDS row stride in floats (pad 64->68: conflict-free)
#define NS_ITERS  28          // enough to push min|lambda|/||Y||_F ~ 1e-4 to sign=+-1

typedef __attribute__((ext_vector_type(2))) float v2f;
typedef __attribute__((ext_vector_type(8))) float v8f;

// Only meaningful to probe builtins in the DEVICE pass; the host pass sees
// amdgcn builtins as aux-target builtins and __has_builtin reports 0 there.
#if defined(__HIP_DEVICE_COMPILE__)
#if !__has_builtin(__builtin_amdgcn_wmma_f32_16x16x4_f32)
#error "device pass: missing __builtin_amdgcn_wmma_f32_16x16x4_f32"
#endif
#endif

// C[16x16] tile at (m0,n0) += A(rows m0..m0+15) * B(cols n0..n0+15), K=64.
// A-frag f32 16x4: lanes 0-15 row m0+lane, v0=K(k0+2h), v1=K(k0+2h+1), h=lane>=16.
// B-frag f32 4x16: lanes hold col n0+(lane&15); v0=row k0+2h, v1=row k0+2h+1.
__device__ __forceinline__ v8f wmma_mm64(const float* __restrict__ A,
                                         const float* __restrict__ B,
                                         int m0, int n0, int lane, v8f acc) {
  const int hh = lane >> 4;
  const int rr = lane & 15;
#pragma unroll
  for (int k0 = 0; k0 < 64; k0 += 4) {
    v2f a, b;
    const float* pa = A + (m0 + rr) * LD + k0 + 2 * hh;
    a.x = pa[0];
    a.y = pa[1];
    const float* pb = B + (k0 + 2 * hh) * LD + n0 + rr;
    b.x = pb[0];
    b.y = pb[LD];
    acc = __builtin_amdgcn_wmma_f32_16x16x4_f32(false, a, false, b,
                                                (short)0, acc, false, false);
  }
  return acc;
}

// D-tile layout: VGPR v holds element [m0 + v + 8*(lane>=16)][n0 + (lane&15)].
__device__ __forceinline__ void store_tile(float* __restrict__ M, v8f acc,
                                           int m0, int n0, int lane) {
  const int hh = lane >> 4;
  const int cc = lane & 15;
#pragma unroll
  for (int v = 0; v < 8; ++v)
    M[(m0 + v + 8 * hh) * LD + n0 + cc] = acc[v];
}

__global__ __launch_bounds__(256) void spd_rectify_sign_kernel(
    const float* __restrict__ x, float* __restrict__ out) {
  __shared__ float sY[64 * LD];   // Y = X - eps*I (kept for final |Y| = S*Y)
  __shared__ float sP[64 * LD];   // Z ping
  __shared__ float sQ[64 * LD];   // Z^2 pong
  __shared__ float s_red;

  const int tid  = threadIdx.x;
  const int lane = tid & 31;
  const int wv   = tid >> 5;              // 8 waves; each owns C tiles (m0,n0),(m0+32,n0)
  const int m0   = (wv >> 2) << 4;        // 0 or 16
  const int n0   = (wv & 3) << 4;         // 0,16,32,48
  const int m1   = m0 + 32;               // 32 or 48
  const int hh   = lane >> 4;
  const int cc   = lane & 15;

  const size_t base = (size_t)blockIdx.x * 4096;
  const float4* __restrict__ xin = (const float4*)(x + base);

  if (tid == 0) s_red = 0.0f;
  __syncthreads();

  // ---- Load X (coalesced b128), form Y = X - eps*I into LDS, accumulate ||Y||_F^2
  float lsum = 0.0f;
#pragma unroll
  for (int j = 0; j < 4; ++j) {
    const int i  = tid + 256 * j;         // float4 index, 1024 total
    const int r  = i >> 4;                // row 0..63
    const int c0 = (i & 15) << 2;         // col 0..60 step 4
    float4 v = xin[i];
    float e0 = v.x - ((r == c0 + 0) ? EPS : 0.0f);
    float e1 = v.y - ((r == c0 + 1) ? EPS : 0.0f);
    float e2 = v.z - ((r == c0 + 2) ? EPS : 0.0f);
    float e3 = v.w - ((r == c0 + 3) ? EPS : 0.0f);
    lsum += e0 * e0 + e1 * e1 + e2 * e2 + e3 * e3;
    *(float4*)&sY[r * LD + c0] = make_float4(e0, e1, e2, e3);
  }
  atomicAdd(&s_red, lsum);                // ds_add_f32
  __syncthreads();

  const float nf2  = s_red;
  const float invc = (nf2 > 1e-30f) ? rsqrtf(nf2) : 0.0f;

  // Z0 = Y / ||Y||_F  (each thread rescales exactly the elements it wrote)
#pragma unroll
  for (int j = 0; j < 4; ++j) {
    const int i  = tid + 256 * j;
    const int r  = i >> 4;
    const int c0 = (i & 15) << 2;
    float4 v = *(float4*)&sY[r * LD + c0];
    *(float4*)&sP[r * LD + c0] =
        make_float4(v.x * invc, v.y * invc, v.z * invc, v.w * invc);
  }

  // ---- Newton-Schulz: Z <- Z*(1.5 I - 0.5 Z^2); spectrum in [-1,1] -> sign(Y)
#pragma unroll 1
  for (int it = 0; it < NS_ITERS; ++it) {
    __syncthreads();                      // sP visible; prior sQ readers done
    v8f t0 = {}, t1 = {};
    t0 = wmma_mm64(sP, sP, m0, n0, lane, t0);
    t1 = wmma_mm64(sP, sP, m1, n0, lane, t1);
    store_tile(sQ, t0, m0, n0, lane);
    store_tile(sQ, t1, m1, n0, lane);
    __syncthreads();                      // sQ = Z^2 visible
    v8f w0 = {}, w1 = {};
    w0 = wmma_mm64(sP, sQ, m0, n0, lane, w0);   // Z * Z^2
    w1 = wmma_mm64(sP, sQ, m1, n0, lane, w1);
#pragma unroll
    for (int v = 0; v < 8; ++v) {
      const int r0 = m0 + v + 8 * hh;
      const int r1 = m1 + v + 8 * hh;
      w0[v] = 1.5f * sP[r0 * LD + n0 + cc] - 0.5f * w0[v];
      w1[v] = 1.5f * sP[r1 * LD + n0 + cc] - 0.5f * w1[v];
    }
    __syncthreads();                      // all reads of sP/sQ complete
    store_tile(sP, w0, m0, n0, lane);
    store_tile(sP, w1, m1, n0, lane);
  }
  __syncthreads();                        // sP = S = sign(Y)

  // ---- out = 0.5*(Y + S*Y) + eps*I, fused epilogue, coalesced-ish b32 stores
  v8f d0 = {}, d1 = {};
  d0 = wmma_mm64(sP, sY, m0, n0, lane, d0);
  d1 = wmma_mm64(sP, sY, m1, n0, lane, d1);

  float* __restrict__ ob = out + base;
#pragma unroll
  for (int v = 0; v < 8; ++v) {
    {
      const int r = m0 + v + 8 * hh, c = n0 + cc;
      const float y = sY[r * LD + c];
      ob[r * 64 + c] = 0.5f * (d0[v] + y) + ((r == c) ? EPS : 0.0f);
    }
    {
      const int r = m1 + v + 8 * hh, c = n0 + cc;
      const float y = sY[r * LD + c];
      ob[r * 64 + c] = 0.5f * (d1[v] + y) + ((r == c) ? EPS : 0.0f);
    }
  }
}

extern "C" void kernel_launch(void* const* d_in, const int* in_sizes, int n_in,
                              void* d_out, int out_size, void* d_ws, size_t ws_size,
                              hipStream_t stream) {
  const float* x = (const float*)d_in[0];
  float* out = (float*)d_out;
  const int B = in_sizes[0] / (64 * 64);  // 8192
  spd_rectify_sign_kernel<<<B, 256, 0, stream>>>(x, out);
}